// Discriminator_82471962018373
// MI455X (gfx1250) — compile-verified
//
#include <hip/hip_runtime.h>
#include <math.h>

typedef __attribute__((ext_vector_type(2))) float v2f;
typedef __attribute__((ext_vector_type(8))) float v8f;

#define N_NODES 50000
#define F_IN    128
#define H_OUT   64
#define NH      (N_NODES * H_OUT)   // 3,200,000

// ---------------------------------------------------------------------------
// deg[n] = 1 (self loop weight), then deg[dst] += w per edge, then rsqrt.
// ---------------------------------------------------------------------------
__global__ void k_init_deg(float* __restrict__ deg) {
    int i = blockIdx.x * blockDim.x + threadIdx.x;
    if (i < N_NODES) deg[i] = 1.0f;
}

__global__ void k_accum_deg(const int* __restrict__ dst,
                            const float* __restrict__ w,
                            float* __restrict__ deg, int E) {
    int e = blockIdx.x * blockDim.x + threadIdx.x;
    if (e < E) atomicAdd(&deg[dst[e]], w[e]);
}

__global__ void k_rsqrt_inplace(float* __restrict__ deg) {
    int i = blockIdx.x * blockDim.x + threadIdx.x;
    if (i < N_NODES) {
        float d = deg[i];
        deg[i] = (d > 0.0f) ? rsqrtf(d) : 0.0f;
    }
}

// ---------------------------------------------------------------------------
// h = x @ W   (50000x128 @ 128x64, fp32, WMMA 16x16x4)
// Block: 128 threads = 4 waves. Block owns a 16-row M tile (3125 blocks,
// 3125*16 == 50000 exactly -> no guards, EXEC stays all-ones for WMMA).
// Wave w owns N columns [16w, 16w+16). W staged in LDS (32 KB).
//
// A (16x4 f32) per ISA 7.12.2: lanes 0-15 -> K={0,1}, lanes 16-31 -> K={2,3},
//   VGPR0=K even, VGPR1=K odd; M = lane&15.
// B (4x16 f32): VGPR0 -> K = 0/2 by lane half, N = lane&15; VGPR1 -> K+1.
// C (16x16 f32, 8 VGPRs): VGPR r -> M = r + 8*(lane>>4), N = lane&15.
// ---------------------------------------------------------------------------
__global__ __launch_bounds__(128) void k_gemm_wmma(const float* __restrict__ x,
                                                   const float* __restrict__ W,
                                                   float* __restrict__ h) {
    __shared__ float lw[F_IN * H_OUT];
    const int tid = threadIdx.x;
    for (int i = tid; i < F_IN * H_OUT; i += 128) lw[i] = W[i];
    __syncthreads();

    const int wave  = tid >> 5;
    const int lane  = tid & 31;
    const int lhalf = lane >> 4;          // 0 or 1
    const int l15   = lane & 15;
    const int mrow  = blockIdx.x * 16 + l15;
    const int ncol  = wave * 16 + l15;
    const int khalf = lhalf << 1;         // 0 or 2

    const float2* xrow = (const float2*)(x + (size_t)mrow * F_IN);

    v8f c = {};
    #pragma unroll 8
    for (int k0 = 0; k0 < F_IN; k0 += 4) {
        const int kk = k0 + khalf;        // even
        float2 av = xrow[kk >> 1];
        v2f a; a.x = av.x; a.y = av.y;
        v2f b; b.x = lw[kk * H_OUT + ncol];
               b.y = lw[(kk + 1) * H_OUT + ncol];
        c = __builtin_amdgcn_wmma_f32_16x16x4_f32(
                /*neg_a=*/false, a, /*neg_b=*/false, b,
                /*c_mod=*/(short)0, c, /*reuse_a=*/false, /*reuse_b=*/false);
    }

    const int mbase = blockIdx.x * 16 + (lhalf << 3);
    #pragma unroll
    for (int r = 0; r < 8; ++r)
        h[(size_t)(mbase + r) * H_OUT + ncol] = c[r];
}

// ---------------------------------------------------------------------------
// out[n,f] = bias[f] + dinv[n]^2 * h[n,f]   (self-loop term, norm = dinv*1*dinv)
// ---------------------------------------------------------------------------
__global__ void k_init_out(const float* __restrict__ h,
                           const float* __restrict__ dinv,
                           const float* __restrict__ bias,
                           float* __restrict__ out) {
    int i = blockIdx.x * blockDim.x + threadIdx.x;
    if (i < NH) {
        int n = i >> 6, f = i & 63;
        float di = dinv[n];
        out[i] = bias[f] + di * di * h[i];
    }
}

// ---------------------------------------------------------------------------
// out[dst,f] += dinv[src]*w*dinv[dst] * h[src,f]; 64 threads per edge.
// ---------------------------------------------------------------------------
__global__ void k_scatter(const int* __restrict__ src,
                          const int* __restrict__ dst,
                          const float* __restrict__ w,
                          const float* __restrict__ h,
                          const float* __restrict__ dinv,
                          float* __restrict__ out, int E) {
    long long idx = (long long)blockIdx.x * blockDim.x + threadIdx.x;
    int e = (int)(idx >> 6);
    int f = (int)(idx & 63);
    if (e < E) {
        int s = src[e], d = dst[e];
        float nrm = dinv[s] * w[e] * dinv[d];
        atomicAdd(&out[(size_t)d * H_OUT + f], h[(size_t)s * H_OUT + f] * nrm);
    }
}

// ---------------------------------------------------------------------------
// Deterministic two-pass reduction: sum relu(out[i]) * fc_w[i]
// ---------------------------------------------------------------------------
#define RED_BLOCKS 1024

__global__ __launch_bounds__(256) void k_reduce1(const float* __restrict__ out,
                                                 const float* __restrict__ fcw,
                                                 float* __restrict__ partial) {
    __shared__ float sm[256];
    float acc = 0.0f;
    for (int i = blockIdx.x * 256 + threadIdx.x; i < NH; i += RED_BLOCKS * 256)
        acc += fmaxf(out[i], 0.0f) * fcw[i];
    sm[threadIdx.x] = acc;
    __syncthreads();
    for (int s = 128; s > 0; s >>= 1) {
        if (threadIdx.x < s) sm[threadIdx.x] += sm[threadIdx.x + s];
        __syncthreads();
    }
    if (threadIdx.x == 0) partial[blockIdx.x] = sm[0];
}

__global__ __launch_bounds__(256) void k_reduce2(const float* __restrict__ partial,
                                                 const float* __restrict__ fcb,
                                                 float* __restrict__ result) {
    __shared__ float sm[256];
    float acc = 0.0f;
    for (int i = threadIdx.x; i < RED_BLOCKS; i += 256) acc += partial[i];
    sm[threadIdx.x] = acc;
    __syncthreads();
    for (int s = 128; s > 0; s >>= 1) {
        if (threadIdx.x < s) sm[threadIdx.x] += sm[threadIdx.x + s];
        __syncthreads();
    }
    if (threadIdx.x == 0) {
        float logit = sm[0] + fcb[0];
        result[0] = 1.0f / (1.0f + expf(-logit));
    }
}

// ---------------------------------------------------------------------------
extern "C" void kernel_launch(void* const* d_in, const int* in_sizes, int n_in,
                              void* d_out, int out_size, void* d_ws, size_t ws_size,
                              hipStream_t stream) {
    const float* x         = (const float*)d_in[0];
    const int*   edge_list = (const int*)  d_in[1];   // [2, E] flat
    const float* edge_attr = (const float*)d_in[2];
    const float* conv_w    = (const float*)d_in[3];   // [128, 64]
    const float* conv_b    = (const float*)d_in[4];   // [64]
    const float* fc_w      = (const float*)d_in[5];   // [N*64]
    const float* fc_b      = (const float*)d_in[6];   // [1]
    float*       result    = (float*)d_out;

    const int E = in_sizes[1] / 2;
    const int* src = edge_list;
    const int* dst = edge_list + E;

    // Workspace layout (bytes)
    char* ws = (char*)d_ws;
    float* h       = (float*)(ws);                          // 12.8 MB
    float* out_ft  = (float*)(ws + (size_t)NH * 4);         // 12.8 MB
    float* dinv    = (float*)(ws + (size_t)NH * 8);         // 200 KB (deg->dinv)
    float* partial = (float*)(ws + (size_t)NH * 8 + 256 * 1024); // 4 KB

    // 1) degree (self loop = 1) + edge weights, then rsqrt in place
    k_init_deg<<<(N_NODES + 255) / 256, 256, 0, stream>>>(dinv);
    k_accum_deg<<<(E + 255) / 256, 256, 0, stream>>>(dst, edge_attr, dinv, E);
    k_rsqrt_inplace<<<(N_NODES + 255) / 256, 256, 0, stream>>>(dinv);

    // 2) h = x @ W via fp32 WMMA (3125 M-tiles x 4 waves of N-tiles)
    k_gemm_wmma<<<N_NODES / 16, 128, 0, stream>>>(x, conv_w, h);

    // 3) out = bias + self-loop term
    k_init_out<<<(NH + 255) / 256, 256, 0, stream>>>(h, dinv, conv_b, out_ft);

    // 4) edge scatter-add (64 threads/edge)
    long long sthreads = (long long)E * 64;
    int sblocks = (int)((sthreads + 255) / 256);
    k_scatter<<<sblocks, 256, 0, stream>>>(src, dst, edge_attr, h, dinv, out_ft, E);

    // 5) relu + dot + bias + sigmoid (deterministic two-pass)
    k_reduce1<<<RED_BLOCKS, 256, 0, stream>>>(out_ft, fc_w, partial);
    k_reduce2<<<1, 256, 0, stream>>>(partial, fc_b, result);
}